// LstmDecoder_12360915877894
// MI455X (gfx1250) — compile-verified
//
#include <hip/hip_runtime.h>
#include <hip/hip_bf16.h>

#define B_ 64
#define H_ 1024
#define E_ 512
#define V_ 32000
#define T_ 64

typedef float v2f __attribute__((ext_vector_type(2)));
typedef float v8f __attribute__((ext_vector_type(8)));

__device__ __forceinline__ float sigmoid_(float x) { return 1.0f / (1.0f + __expf(-x)); }
__device__ __forceinline__ float tanh_(float x)    { return 2.0f / (1.0f + __expf(-2.0f * x)) - 1.0f; }

// ---------------------------------------------------------------------------
// Init: copy immutable inputs h,c into ping-pong state buffers; tok = SOS(0).
// ---------------------------------------------------------------------------
__global__ void init_state_kernel(const float* __restrict__ h, const float* __restrict__ c,
                                  float* __restrict__ h0, float* __restrict__ c0,
                                  float* __restrict__ h1, float* __restrict__ c1,
                                  int* __restrict__ tok) {
    int i = blockIdx.x * blockDim.x + threadIdx.x;
    if (i < B_ * H_) {
        h0[i] = h[i];            // layer 0 state
        h1[i] = h[B_ * H_ + i];  // layer 1 state
        c0[i] = c[i];
        c1[i] = c[B_ * H_ + i];
    }
    if (i < B_) tok[i] = 0;  // SOS token id
}

// ---------------------------------------------------------------------------
// One LSTM layer for the whole batch.  gates = x@w_ih.T + h@w_hh.T + b.
// Each wave owns a 16-wide N-slice of the hidden dim for the ENTIRE batch
// (4 M-tiles): every weight fragment is loaded once and reused by 4 WMMAs,
// minimizing weight bandwidth (the roofline bottleneck).
// If x == nullptr the A operand is gathered from emb[tok[row]] (layer 0).
// Grid: 8 blocks x 256 threads = 64 waves = 1024/16 N-tiles.
// ---------------------------------------------------------------------------
__global__ __launch_bounds__(256) void lstm_layer_kernel(
    const float* __restrict__ x, const int* __restrict__ tok, const float* __restrict__ emb,
    const float* __restrict__ h_in, const float* __restrict__ c_in,
    const float* __restrict__ w_ih, const float* __restrict__ w_hh,
    const float* __restrict__ b_ih, const float* __restrict__ b_hh,
    float* __restrict__ h_out, float* __restrict__ c_out, int Ein)
{
    const int tid  = threadIdx.x;
    const int lane = tid & 31;
    const int wave = tid >> 5;
    const int nt   = blockIdx.x * 8 + wave;   // 0..63
    const int n0   = nt * 16;                 // hidden tile origin
    const int ncol = lane & 15;
    const int half = lane >> 4;
    const int koff = half * 2;                // A/B fragment K-offset per half-wave

    // A operands: one 16x4 fragment per M-tile (4 tiles cover batch 64).
    const float* ab[4];
#pragma unroll
    for (int mt = 0; mt < 4; ++mt) {
        const int arow = mt * 16 + ncol;
        ab[mt] = (x ? (x + (size_t)arow * Ein)
                    : (emb + (size_t)tok[arow] * Ein)) + koff;
    }

    // B operands: weight rows are exactly the Bᵀ columns we need (contiguous K).
    const float* wi[4];
    const float* wh[4];
#pragma unroll
    for (int g = 0; g < 4; ++g) {
        wi[g] = w_ih + (size_t)(g * H_ + n0 + ncol) * Ein + koff;
        wh[g] = w_hh + (size_t)(g * H_ + n0 + ncol) * H_  + koff;
    }

    v8f acc[4][4];  // [gate][m-tile]
#pragma unroll
    for (int g = 0; g < 4; ++g)
#pragma unroll
        for (int mt = 0; mt < 4; ++mt) acc[g][mt] = {};

    // Phase 1: K over Ein (input weights)
#pragma unroll 2
    for (int k = 0; k < Ein; k += 4) {
        v2f a[4];
#pragma unroll
        for (int mt = 0; mt < 4; ++mt) a[mt] = *(const v2f*)(ab[mt] + k);
#pragma unroll
        for (int g = 0; g < 4; ++g) {
            v2f b = *(const v2f*)(wi[g] + k);
#pragma unroll
            for (int mt = 0; mt < 4; ++mt)
                acc[g][mt] = __builtin_amdgcn_wmma_f32_16x16x4_f32(
                    false, a[mt], false, b, (short)0, acc[g][mt], false, false);
        }
    }

    // Phase 2: K over H (recurrent weights)
    const float* hb[4];
#pragma unroll
    for (int mt = 0; mt < 4; ++mt)
        hb[mt] = h_in + (size_t)(mt * 16 + ncol) * H_ + koff;
#pragma unroll 2
    for (int k = 0; k < H_; k += 4) {
        v2f a[4];
#pragma unroll
        for (int mt = 0; mt < 4; ++mt) a[mt] = *(const v2f*)(hb[mt] + k);
#pragma unroll
        for (int g = 0; g < 4; ++g) {
            v2f b = *(const v2f*)(wh[g] + k);
#pragma unroll
            for (int mt = 0; mt < 4; ++mt)
                acc[g][mt] = __builtin_amdgcn_wmma_f32_16x16x4_f32(
                    false, a[mt], false, b, (short)0, acc[g][mt], false, false);
        }
    }

    // Epilogue: bias + activations + cell update.  C/D layout: VGPR v, lanes
    // 0-15 -> M = v, lanes 16-31 -> M = v+8; N = lane&15.
    float bias[4];
#pragma unroll
    for (int g = 0; g < 4; ++g)
        bias[g] = b_ih[g * H_ + n0 + ncol] + b_hh[g * H_ + n0 + ncol];

#pragma unroll
    for (int mt = 0; mt < 4; ++mt) {
        const int rbase = mt * 16 + half * 8;
#pragma unroll
        for (int v = 0; v < 8; ++v) {
            const size_t idx = (size_t)(rbase + v) * H_ + (n0 + ncol);
            float iv = sigmoid_(acc[0][mt][v] + bias[0]);
            float fv = sigmoid_(acc[1][mt][v] + bias[1]);
            float gv = tanh_(acc[2][mt][v] + bias[2]);
            float ov = sigmoid_(acc[3][mt][v] + bias[3]);
            float cn = fv * c_in[idx] + iv * gv;
            c_out[idx] = cn;
            h_out[idx] = ov * tanh_(cn);
        }
    }
}

// ---------------------------------------------------------------------------
// FC head: logits[b, t, :] = h1 @ fc_w.T + fc_b, written straight to d_out.
// Each wave owns one 16-wide vocab slice for the entire batch: fc_w (128 MB,
// the dominant traffic) is read exactly once per step.
// Grid: 250 blocks x 8 waves = 2000 N-tiles = 32000/16.
// ---------------------------------------------------------------------------
__global__ __launch_bounds__(256) void fc_logits_kernel(
    const float* __restrict__ h, const float* __restrict__ fc_w,
    const float* __restrict__ fc_b, float* __restrict__ out, int t)
{
    const int tid  = threadIdx.x;
    const int lane = tid & 31;
    const int wave = tid >> 5;
    const int nt   = blockIdx.x * 8 + wave;   // 0..1999
    const int n0   = nt * 16;
    const int ncol = lane & 15;
    const int half = lane >> 4;
    const int koff = half * 2;

    const float* ab[4];
#pragma unroll
    for (int mt = 0; mt < 4; ++mt)
        ab[mt] = h + (size_t)(mt * 16 + ncol) * H_ + koff;
    const float* wbase = fc_w + (size_t)(n0 + ncol) * H_ + koff;

    v8f acc[4];
#pragma unroll
    for (int mt = 0; mt < 4; ++mt) acc[mt] = {};

#pragma unroll 2
    for (int k = 0; k < H_; k += 4) {
        v2f b = *(const v2f*)(wbase + k);
#pragma unroll
        for (int mt = 0; mt < 4; ++mt) {
            v2f a = *(const v2f*)(ab[mt] + k);
            acc[mt] = __builtin_amdgcn_wmma_f32_16x16x4_f32(
                false, a, false, b, (short)0, acc[mt], false, false);
        }
    }

    const float bias = fc_b[n0 + ncol];
#pragma unroll
    for (int mt = 0; mt < 4; ++mt) {
        const int rbase = mt * 16 + half * 8;
#pragma unroll
        for (int v = 0; v < 8; ++v) {
            const size_t o = ((size_t)(rbase + v) * T_ + t) * V_ + (n0 + ncol);
            out[o] = acc[mt][v] + bias;
        }
    }
}

// ---------------------------------------------------------------------------
// Greedy argmax per batch row (first-index tie-break, matching jnp.argmax).
// One block per batch row.
// ---------------------------------------------------------------------------
__global__ __launch_bounds__(256) void argmax_kernel(const float* __restrict__ logits,
                                                     int t, int* __restrict__ tok)
{
    const int b   = blockIdx.x;
    const int tid = threadIdx.x;
    const float* row = logits + ((size_t)b * T_ + t) * V_;

    float best = -3.402823466e+38f;
    int   bidx = V_;
    for (int v = tid; v < V_; v += 256) {
        float x = row[v];
        if (x > best) { best = x; bidx = v; }   // ascending v => first occurrence
    }

    __shared__ float sv[256];
    __shared__ int   si[256];
    sv[tid] = best; si[tid] = bidx;
    __syncthreads();
#pragma unroll
    for (int s = 128; s > 0; s >>= 1) {
        if (tid < s) {
            if (sv[tid + s] > sv[tid] ||
                (sv[tid + s] == sv[tid] && si[tid + s] < si[tid])) {
                sv[tid] = sv[tid + s];
                si[tid] = si[tid + s];
            }
        }
        __syncthreads();
    }
    if (tid == 0) tok[b] = si[0];
}

// ---------------------------------------------------------------------------
extern "C" void kernel_launch(void* const* d_in, const int* in_sizes, int n_in,
                              void* d_out, int out_size, void* d_ws, size_t ws_size,
                              hipStream_t stream) {
    const float* h     = (const float*)d_in[0];
    const float* c     = (const float*)d_in[1];
    const float* emb   = (const float*)d_in[2];
    const float* w_ih0 = (const float*)d_in[3];
    const float* w_hh0 = (const float*)d_in[4];
    const float* b_ih0 = (const float*)d_in[5];
    const float* b_hh0 = (const float*)d_in[6];
    const float* w_ih1 = (const float*)d_in[7];
    const float* w_hh1 = (const float*)d_in[8];
    const float* b_ih1 = (const float*)d_in[9];
    const float* b_hh1 = (const float*)d_in[10];
    const float* fc_w  = (const float*)d_in[11];
    const float* fc_b  = (const float*)d_in[12];
    float* out = (float*)d_out;

    // Workspace layout: tok[64] then 8 ping-pong state buffers of B*H floats.
    int*   tok  = (int*)d_ws;
    float* base = (float*)((char*)d_ws + 256);
    const size_t BH = (size_t)B_ * H_;
    float* h0[2] = { base + 0 * BH, base + 1 * BH };
    float* c0[2] = { base + 2 * BH, base + 3 * BH };
    float* h1[2] = { base + 4 * BH, base + 5 * BH };
    float* c1[2] = { base + 6 * BH, base + 7 * BH };

    init_state_kernel<<<(B_ * H_ + 255) / 256, 256, 0, stream>>>(
        h, c, h0[0], c0[0], h1[0], c1[0], tok);

    for (int t = 0; t < T_; ++t) {
        const int r = t & 1;       // read buffer
        const int w = 1 - r;       // write buffer
        // layer 0: embedding gather fused as the A operand
        lstm_layer_kernel<<<8, 256, 0, stream>>>(
            nullptr, tok, emb, h0[r], c0[r],
            w_ih0, w_hh0, b_ih0, b_hh0, h0[w], c0[w], E_);
        // layer 1: input is layer 0's fresh hidden state
        lstm_layer_kernel<<<8, 256, 0, stream>>>(
            h0[w], nullptr, nullptr, h1[r], c1[r],
            w_ih1, w_hh1, b_ih1, b_hh1, h1[w], c1[w], H_);
        // FC head -> logits into d_out[:, t, :]
        fc_logits_kernel<<<250, 256, 0, stream>>>(h1[w], fc_w, fc_b, out, t);
        // greedy next token
        argmax_kernel<<<B_, 256, 0, stream>>>(out, t, tok);
    }
}